// VQVAE_Probe_23742579212382
// MI455X (gfx1250) — compile-verified
//
#include <hip/hip_runtime.h>

typedef __attribute__((ext_vector_type(2))) float v2f;
typedef __attribute__((ext_vector_type(4))) float v4f;
typedef __attribute__((ext_vector_type(8))) float v8f;

#define WMMA_F32(a, b, c) \
  __builtin_amdgcn_wmma_f32_16x16x4_f32(false, (a), false, (b), (short)0, (c), false, false)

// Async global->LDS copy (16B per lane), tracked by ASYNCcnt.
__device__ __forceinline__ void async_copy_b128(unsigned lds_byte_off,
                                                const float* gaddr) {
  asm volatile("global_load_async_to_lds_b128 %0, %1, off"
               :
               : "v"(lds_byte_off), "v"(gaddr)
               : "memory");
}
__device__ __forceinline__ void wait_async0() {
  asm volatile("s_wait_asynccnt 0x0" ::: "memory");
}
__device__ __forceinline__ unsigned lds_off32(const void* p) {
  // LDS aperture: flat address low 32 bits are the LDS byte offset.
  return (unsigned)(unsigned long long)p;
}

static constexpr int Bn    = 16384;
static constexpr int D     = 512;
static constexpr int H     = 64;
static constexpr int V     = 10000;
static constexpr int NROOT = 5000;
static constexpr int NSFX  = 2000;
static constexpr int NORD  = 1000;
static constexpr int KORD  = 6;
static constexpr int SUF   = H + KORD * H;   // 448

// ---------------------------------------------------------------------------
// Per-row squared norms of a codebook: out[i] = ||mat[i,:]||^2
// ---------------------------------------------------------------------------
__global__ void norms_kernel(const float* __restrict__ mat,
                             float* __restrict__ out, int N, int Dm) {
  int i = blockIdx.x * blockDim.x + threadIdx.x;
  if (i >= N) return;
  const float* r = mat + (size_t)i * Dm;
  float s = 0.f;
  for (int k = 0; k < Dm; ++k) s += r[k] * r[k];
  out[i] = s;
}

// ---------------------------------------------------------------------------
// logits = fhs @ W_out + b_out   [16384x512] @ [512x10000]
// Block tile: 64(M) x 128(N). 8 waves = 2(M) x 4(N); each wave has a 2x2
// accumulator grid. A is staged per K-slab with double-buffered ASYNC
// global->LDS copies overlapped with the WMMA loop.
// ---------------------------------------------------------------------------
__global__ __launch_bounds__(256) void logits_kernel(
    const float* __restrict__ fhs, const float* __restrict__ Wout,
    const float* __restrict__ bout, float* __restrict__ out) {
  constexpr int KSLAB = 128;
  constexpr int LDB   = KSLAB + 4;       // 132: conflict-free, b128-aligned
  __shared__ float ldsA[2][64 * LDB];    // ~68 KB double buffer

  const int tid  = threadIdx.x;
  const int lane = tid & 31;
  const int wave = tid >> 5;
  const int wm   = wave & 1;   // M half of the block
  const int wn   = wave >> 1;  // N quarter of the block
  const int m0   = blockIdx.x * 64;
  const int lm   = lane & 15;
  const int koff = (lane >> 4) * 2;
  const int hsel = (lane >> 4) << 3;

  const unsigned ldsBase = lds_off32(&ldsA[0][0]);

  int  col[2];
  bool valid[2];
#pragma unroll
  for (int j = 0; j < 2; ++j) {
    int tile = blockIdx.y * 8 + wn * 2 + j;
    valid[j] = tile < (V / 16);
    int c    = tile * 16 + lm;
    col[j]   = c < V ? c : V - 1;
  }

  v8f acc[2][2];
#pragma unroll
  for (int i = 0; i < 2; ++i)
#pragma unroll
    for (int j = 0; j < 2; ++j) acc[i][j] = {};

  auto prefetch_slab = [&](int buf, int k0) {
    unsigned base = ldsBase + (unsigned)(buf * 64 * LDB * 4);
    for (int u = tid; u < 64 * (KSLAB / 4); u += 256) {
      int r  = u >> 5;           // 32 float4 per row
      int c4 = (u & 31) << 2;
      async_copy_b128(base + (unsigned)((r * LDB + c4) * 4),
                      fhs + (size_t)(m0 + r) * D + k0 + c4);
    }
  };

  prefetch_slab(0, 0);
  wait_async0();
  __syncthreads();

  int buf = 0;
  for (int k0 = 0; k0 < D; k0 += KSLAB, buf ^= 1) {
    if (k0 + KSLAB < D) prefetch_slab(buf ^ 1, k0 + KSLAB);

    const float* As = &ldsA[buf][0];
    for (int kk = 0; kk < KSLAB; kk += 4) {
      v2f a0 = *(const v2f*)&As[((wm * 2 + 0) * 16 + lm) * LDB + kk + koff];
      v2f a1 = *(const v2f*)&As[((wm * 2 + 1) * 16 + lm) * LDB + kk + koff];
      const float* w0 = Wout + (size_t)(k0 + kk + koff) * V;
#pragma unroll
      for (int j = 0; j < 2; ++j) {
        v2f b;
        b.x = w0[col[j]];
        b.y = w0[col[j] + V];
        acc[0][j] = WMMA_F32(a0, b, acc[0][j]);
        acc[1][j] = WMMA_F32(a1, b, acc[1][j]);
      }
    }
    wait_async0();     // next slab landed
    __syncthreads();   // everyone done reading current slab
  }

#pragma unroll
  for (int j = 0; j < 2; ++j) {
    if (!valid[j]) continue;
    int   c    = col[j];
    float bias = bout[c];
#pragma unroll
    for (int i = 0; i < 2; ++i) {
#pragma unroll
      for (int v = 0; v < 8; ++v) {
        int row = m0 + (wm * 2 + i) * 16 + v + hsel;
        out[(size_t)row * V + c] = acc[i][j][v] + bias;
      }
    }
  }
}

// ---------------------------------------------------------------------------
// Root VQ: one wave owns 64 rows (4 M sub-tiles) with the full A stripe in
// LDS (64x516 floats ~ 132 KB), staged with async global->LDS copies.
// Each codebook fragment (one global b64) feeds 4 WMMAs.
// argmin of ||e||^2 - 2 x.e per row, then b128 gather of winning rows.
// ---------------------------------------------------------------------------
__global__ __launch_bounds__(32) void rootvq_kernel(
    const float* __restrict__ fhs, const float* __restrict__ cb,
    const float* __restrict__ norms, float* __restrict__ outq) {
  constexpr int LDA = D + 4;          // 516
  __shared__ float ldsA[64 * LDA];    // ~132 KB
  __shared__ float redV[4 * 8 * 32];
  __shared__ int   redI[4 * 8 * 32];
  __shared__ int   bestIdx[64];

  const int lane = threadIdx.x;
  const int m0   = blockIdx.x * 64;
  const int lm   = lane & 15;
  const int koff = (lane >> 4) * 2;

  // Stage 64 rows of fhs via async b128 copies (ASYNCcnt-throttled by HW).
  const unsigned ldsBase = lds_off32(&ldsA[0]);
  for (int u = lane; u < 64 * (D / 4); u += 32) {
    int r  = u >> 7;            // 128 float4 per row
    int c4 = (u & 127) << 2;
    async_copy_b128(ldsBase + (unsigned)((r * LDA + c4) * 4),
                    fhs + (size_t)(m0 + r) * D + c4);
  }
  wait_async0();
  __syncthreads();

  float minv[4][8];
  int   mini[4][8];
#pragma unroll
  for (int s = 0; s < 4; ++s)
#pragma unroll
    for (int v = 0; v < 8; ++v) { minv[s][v] = 3.4e38f; mini[s][v] = 0; }

  const int NT = (NROOT + 15) / 16;  // 313 (tail clamped)
  for (int nt = 0; nt < NT; ++nt) {
    int n  = nt * 16 + lm;
    int nc = n < NROOT ? n : NROOT - 1;
    const float* brow = cb + (size_t)nc * D;

    v8f acc[4];
#pragma unroll
    for (int s = 0; s < 4; ++s) acc[s] = {};

    for (int k = 0; k < D; k += 4) {
      v2f b = *(const v2f*)(brow + k + koff);
#pragma unroll
      for (int s = 0; s < 4; ++s) {
        v2f a = *(const v2f*)&ldsA[(s * 16 + lm) * LDA + k + koff];
        acc[s] = WMMA_F32(a, b, acc[s]);
      }
    }
    if (n < NROOT) {
      float nr = norms[n];
#pragma unroll
      for (int s = 0; s < 4; ++s)
#pragma unroll
        for (int v = 0; v < 8; ++v) {
          float dist = nr - 2.f * acc[s][v];
          if (dist < minv[s][v]) { minv[s][v] = dist; mini[s][v] = n; }
        }
    }
  }

#pragma unroll
  for (int s = 0; s < 4; ++s)
#pragma unroll
    for (int v = 0; v < 8; ++v) {
      redV[(s * 8 + v) * 32 + lane] = minv[s][v];
      redI[(s * 8 + v) * 32 + lane] = mini[s][v];
    }
  __syncthreads();
  if (lane < 16) {
    int v = lane & 7, half = lane >> 3;
    for (int s = 0; s < 4; ++s) {
      float bv = 3.4e38f;
      int   bi = 0;
      for (int l = half * 16; l < half * 16 + 16; ++l) {
        float x  = redV[(s * 8 + v) * 32 + l];
        int   ix = redI[(s * 8 + v) * 32 + l];
        if (x < bv || (x == bv && ix < bi)) { bv = x; bi = ix; }
      }
      bestIdx[s * 16 + lane] = bi;
    }
  }
  __syncthreads();

  for (int r = 0; r < 64; ++r) {
    const float* src = cb + (size_t)bestIdx[r] * D;
    float*       dst = outq + (size_t)(m0 + r) * D;
    for (int k4 = lane * 4; k4 < D; k4 += 128)
      *(v4f*)&dst[k4] = *(const v4f*)&src[k4];
  }
}

// ---------------------------------------------------------------------------
// Fused projection + VQ (sfx head via grid.y=1, 6 ordinal heads via grid.y=6).
// proj = fhs @ W[k] + b[k] -> [16x64] in LDS, then VQ vs cb[k] [Ncb x 64].
// ---------------------------------------------------------------------------
__global__ __launch_bounds__(32) void projvq_kernel(
    const float* __restrict__ fhs, const float* __restrict__ Wall,
    const float* __restrict__ ball, const float* __restrict__ cball,
    const float* __restrict__ normsall, int Ncb,
    float* __restrict__ outSuffix, int outOff) {
  constexpr int LDA = D + 4;  // 516
  constexpr int LDP = H + 4;  // 68
  __shared__ float ldsA[16 * LDA];
  __shared__ float ldsP[16 * LDP];
  __shared__ float redV[8 * 32];
  __shared__ int   redI[8 * 32];
  __shared__ int   bestIdx[16];

  const int lane = threadIdx.x;
  const int kb   = blockIdx.y;
  const float* W     = Wall + (size_t)kb * D * H;
  const float* bias  = ball + (size_t)kb * H;
  const float* cb    = cball + (size_t)kb * Ncb * H;
  const float* norms = normsall + (size_t)kb * Ncb;
  const int colBase  = outOff + kb * H;

  const int m0   = blockIdx.x * 16;
  const int lm   = lane & 15;
  const int koff = (lane >> 4) * 2;

  const unsigned ldsBase = lds_off32(&ldsA[0]);
  for (int u = lane; u < 16 * (D / 4); u += 32) {
    int r  = u >> 7;
    int c4 = (u & 127) << 2;
    async_copy_b128(ldsBase + (unsigned)((r * LDA + c4) * 4),
                    fhs + (size_t)(m0 + r) * D + c4);
  }
  wait_async0();
  __syncthreads();

  // Phase 1: projection [16x512] @ [512x64]
  v8f acc[4];
#pragma unroll
  for (int t = 0; t < 4; ++t) acc[t] = {};
  for (int k = 0; k < D; k += 4) {
    v2f a = *(const v2f*)&ldsA[lm * LDA + k + koff];
    const float* w0 = W + (size_t)(k + koff) * H;
#pragma unroll
    for (int t = 0; t < 4; ++t) {
      int h = t * 16 + lm;
      v2f b;
      b.x = w0[h];
      b.y = w0[h + H];
      acc[t] = WMMA_F32(a, b, acc[t]);
    }
  }
#pragma unroll
  for (int t = 0; t < 4; ++t) {
    int   h  = t * 16 + lm;
    float bv = bias[h];
#pragma unroll
    for (int v = 0; v < 8; ++v) {
      int row = v + ((lane >> 4) << 3);
      ldsP[row * LDP + h] = acc[t][v] + bv;
    }
  }
  __syncthreads();

  // Phase 2: VQ against cb [Ncb x 64]
  float minv[8];
  int   mini[8];
#pragma unroll
  for (int v = 0; v < 8; ++v) { minv[v] = 3.4e38f; mini[v] = 0; }
  const int NT = (Ncb + 15) / 16;
  for (int nt = 0; nt < NT; ++nt) {
    int n  = nt * 16 + lm;
    int nc = n < Ncb ? n : Ncb - 1;
    const float* brow = cb + (size_t)nc * H;
    v8f dacc = {};
    for (int k = 0; k < H; k += 4) {
      v2f a = *(const v2f*)&ldsP[lm * LDP + k + koff];
      v2f b = *(const v2f*)(brow + k + koff);
      dacc = WMMA_F32(a, b, dacc);
    }
    if (n < Ncb) {
      float nr = norms[n];
#pragma unroll
      for (int v = 0; v < 8; ++v) {
        float dist = nr - 2.f * dacc[v];
        if (dist < minv[v]) { minv[v] = dist; mini[v] = n; }
      }
    }
  }

#pragma unroll
  for (int v = 0; v < 8; ++v) {
    redV[v * 32 + lane] = minv[v];
    redI[v * 32 + lane] = mini[v];
  }
  __syncthreads();
  if (lane < 16) {
    int   v = lane & 7, half = lane >> 3;
    float bv = 3.4e38f;
    int   bi = 0;
    for (int l = half * 16; l < half * 16 + 16; ++l) {
      float x  = redV[v * 32 + l];
      int   ix = redI[v * 32 + l];
      if (x < bv || (x == bv && ix < bi)) { bv = x; bi = ix; }
    }
    bestIdx[lane] = bi;
  }
  __syncthreads();

  for (int r = 0; r < 16; ++r) {
    const float* src = cb + (size_t)bestIdx[r] * H;
    float*       dst = outSuffix + (size_t)(m0 + r) * SUF + colBase;
    for (int h4 = lane * 4; h4 < H; h4 += 128)
      *(v4f*)&dst[h4] = *(const v4f*)&src[h4];
  }
}

// ---------------------------------------------------------------------------
extern "C" void kernel_launch(void* const* d_in, const int* in_sizes, int n_in,
                              void* d_out, int out_size, void* d_ws,
                              size_t ws_size, hipStream_t stream) {
  const float* fhs     = (const float*)d_in[0];
  const float* cb_root = (const float*)d_in[1];
  const float* W_sfx   = (const float*)d_in[2];
  const float* b_sfx   = (const float*)d_in[3];
  const float* cb_sfx  = (const float*)d_in[4];
  const float* W_ord   = (const float*)d_in[5];
  const float* b_ord   = (const float*)d_in[6];
  const float* cb_ord  = (const float*)d_in[7];
  const float* W_out   = (const float*)d_in[8];
  const float* b_out   = (const float*)d_in[9];

  float* out_logits = (float*)d_out;                   // [16384 x 10000]
  float* out_rootq  = out_logits + (size_t)Bn * V;     // [16384 x 512]
  float* out_suffix = out_rootq + (size_t)Bn * D;      // [16384 x 448]

  float* nrm_root = (float*)d_ws;      // 5000
  float* nrm_sfx  = nrm_root + NROOT;  // 2000
  float* nrm_ord  = nrm_sfx + NSFX;    // 6000

  norms_kernel<<<dim3((NROOT + 255) / 256), 256, 0, stream>>>(cb_root, nrm_root, NROOT, D);
  norms_kernel<<<dim3((NSFX + 255) / 256), 256, 0, stream>>>(cb_sfx, nrm_sfx, NSFX, H);
  norms_kernel<<<dim3((KORD * NORD + 255) / 256), 256, 0, stream>>>(cb_ord, nrm_ord, KORD * NORD, H);

  // logits GEMM: 64x128 block tiles, double-buffered async A staging.
  logits_kernel<<<dim3(Bn / 64, ((V / 16) + 7) / 8), 256, 0, stream>>>(
      fhs, W_out, b_out, out_logits);

  // Root VQ: 64 rows per wave, codebook fragment reused 4x.
  rootvq_kernel<<<dim3(Bn / 64), 32, 0, stream>>>(fhs, cb_root, nrm_root, out_rootq);

  // Suffix head (cols 0..63 of suffix_concat)
  projvq_kernel<<<dim3(Bn / 16, 1), 32, 0, stream>>>(
      fhs, W_sfx, b_sfx, cb_sfx, nrm_sfx, NSFX, out_suffix, 0);

  // 6 ordinal heads (cols 64 + k*64 .. of suffix_concat)
  projvq_kernel<<<dim3(Bn / 16, KORD), 32, 0, stream>>>(
      fhs, W_ord, b_ord, cb_ord, nrm_ord, NORD, out_suffix, H);
}